// GraphConvolution_30073361007326
// MI455X (gfx1250) — compile-verified
//
#include <hip/hip_runtime.h>

// GCN layer: h = concat_s(segment_sum(x[src_s]*w_s, dst_s)); out = h @ W
// x:[20000,256] f32, edges:[3,320000], W:[768,256] f32, out:[20000,256] f32

#define N_NODES   20000
#define D_FEAT    256
#define N_SUPPORT 3
#define N_EDGES   320000
#define OUT_DIM   256
#define H_DIM     (N_SUPPORT * D_FEAT)   // 768

typedef __attribute__((ext_vector_type(2))) float v2f;
typedef __attribute__((ext_vector_type(8))) float v8f;

// ---------------------------------------------------------------- zero h ----
__global__ __launch_bounds__(256) void zero_h(float4* __restrict__ h, int n4) {
  int i = blockIdx.x * blockDim.x + threadIdx.x;
  if (i < n4) h[i] = float4{0.f, 0.f, 0.f, 0.f};
}

// ------------------------------------------------------- edge scatter-add ----
// 64 threads per edge; thread t gathers x[src][4t..4t+3] (float4, coalesced
// across the 1KB feature row) and atomically accumulates into h[dst].
__global__ __launch_bounds__(256) void scatter_edges(
    const float* __restrict__ x, const int* __restrict__ esrc,
    const int* __restrict__ edst, const float* __restrict__ ew,
    float* __restrict__ h) {
  long gid = (long)blockIdx.x * blockDim.x + threadIdx.x;
  int  chunk = (int)(gid & 63);           // feature chunk (4 floats each)
  long e = gid >> 6;                      // flat edge index in [0, 3*320000)
  if (e >= (long)N_SUPPORT * N_EDGES) return;
  int s   = (int)(e / N_EDGES);           // support id
  int src = esrc[e];
  int dst = edst[e];
  float w = ew[e];
  float4 xv = ((const float4*)(x + (long)src * D_FEAT))[chunk];
  float* hp = h + (long)dst * H_DIM + s * D_FEAT + chunk * 4;
  atomicAdd(hp + 0, xv.x * w);
  atomicAdd(hp + 1, xv.y * w);
  atomicAdd(hp + 2, xv.z * w);
  atomicAdd(hp + 3, xv.w * w);
}

// ------------------------------------------------------------ WMMA GEMM ----
// out[20000,256] = h[20000,768] @ W[768,256], fp32 via V_WMMA_F32_16X16X4_F32.
// One wave32 computes one 16x16 output tile; K-loop steps by 4.
// Fragment layouts (CDNA5 ISA 7.12.2, 32-bit):
//   A 16x4 : lane L holds M=L&15; VGPR0/1 = K = (L>>4)*2 + {0,1}
//   B 4x16 : lane L holds N=L&15; VGPR0/1 = K = (L>>4)*2 + {0,1}
//   C 16x16: lane L, VGPR j -> row = (L>>4)*8 + j, col = L&15
__global__ __launch_bounds__(256) void gemm_wmma(
    const float* __restrict__ h, const float* __restrict__ Wm,
    float* __restrict__ out) {
  int wave = (int)((blockIdx.x * blockDim.x + threadIdx.x) >> 5);
  int lane = threadIdx.x & 31;
  int tile_m = wave >> 4;                 // 0..1249  (20000/16 row tiles)
  int tile_n = wave & 15;                 // 0..15    (256/16 col tiles)
  int half   = lane >> 4;                 // K-pair selector
  int mrow   = tile_m * 16 + (lane & 15);
  int ncol   = tile_n * 16 + (lane & 15);

  const float* arow = h  + (long)mrow * H_DIM + half * 2;       // contiguous f32x2
  const float* bcol = Wm + (long)(half * 2) * OUT_DIM + ncol;   // stride OUT_DIM

  v8f c = {};
  #pragma unroll 4
  for (int k = 0; k < H_DIM; k += 4) {
    v2f a = *(const v2f*)(arow + k);
    v2f b;
    b.x = bcol[(long)k * OUT_DIM];
    b.y = bcol[(long)k * OUT_DIM + OUT_DIM];
    c = __builtin_amdgcn_wmma_f32_16x16x4_f32(
        /*neg_a=*/false, a, /*neg_b=*/false, b,
        /*c_mod=*/(short)0, c, /*reuse_a=*/false, /*reuse_b=*/false);
  }

  float* op = out + (long)(tile_m * 16 + half * 8) * OUT_DIM + ncol;
  #pragma unroll
  for (int j = 0; j < 8; ++j) op[(long)j * OUT_DIM] = c[j];
}

// ---------------------------------------------------------------- launch ----
extern "C" void kernel_launch(void* const* d_in, const int* in_sizes, int n_in,
                              void* d_out, int out_size, void* d_ws, size_t ws_size,
                              hipStream_t stream) {
  const float* x    = (const float*)d_in[0];
  const int*   esrc = (const int*)  d_in[1];
  const int*   edst = (const int*)  d_in[2];
  const float* ew   = (const float*)d_in[3];
  const float* Wm   = (const float*)d_in[4];
  float*       out  = (float*)d_out;
  float*       hbuf = (float*)d_ws;       // 20000*768*4 = 61.44 MB scratch

  // 1) zero the aggregation buffer (atomics accumulate into it)
  int n4 = N_NODES * H_DIM / 4;
  zero_h<<<(n4 + 255) / 256, 256, 0, stream>>>((float4*)hbuf, n4);

  // 2) scatter: 960000 edges * 64 threads = 61.44M threads = 240000 blocks
  long nthr = (long)N_SUPPORT * N_EDGES * 64;
  scatter_edges<<<(int)(nthr / 256), 256, 0, stream>>>(x, esrc, edst, ew, hbuf);

  // 3) GEMM: 1250*16 = 20000 tiles, 8 waves/block -> 2500 blocks (exact fit)
  gemm_wmma<<<(N_NODES / 16) * (OUT_DIM / 16) / 8, 256, 0, stream>>>(hbuf, Wm, out);
}